// TrillNet_74217034874924
// MI455X (gfx1250) — compile-verified
//
#include <hip/hip_runtime.h>
#include <hip/hip_bf16.h>

typedef float v2f __attribute__((ext_vector_type(2)));
typedef float v8f __attribute__((ext_vector_type(8)));

#define Bn 8
#define Hn 32
#define Sn 64
#define Cn 4096
#define Dn 128
#define ROWS (Bn * Cn)          // 32768 rows of E / Y / Eg
#define MASK_WORDS (Cn / 32)    // 128 words per adjacency row

// ---------------------------------------------------------------------------
// 1) Build binary adjacency bitmask + deduplicated column counts.
//    One thread per (b, h, s) transition pair.
__global__ void build_adj_kernel(const int* __restrict__ history,
                                 unsigned* __restrict__ mask,
                                 int* __restrict__ colcnt) {
    int t = blockIdx.x * blockDim.x + threadIdx.x;
    const int NT = Bn * Hn * (Sn - 1);
    if (t >= NT) return;
    int b = t / (Hn * (Sn - 1));
    int r = t % (Hn * (Sn - 1));
    int h = r / (Sn - 1);
    int s = r % (Sn - 1);
    const int* hp = history + ((b * Hn + h) * Sn);
    int h1 = hp[s];
    int h2 = hp[s + 1];
    if (h1 != h2) {
        unsigned bit = 1u << (h2 & 31);
        unsigned* w = mask + ((size_t)(b * Cn + h1) * MASK_WORDS) + (h2 >> 5);
        unsigned old = atomicOr(w, bit);
        if (!(old & bit)) atomicAdd(colcnt + b * Cn + h2, 1); // first setter counts
    }
}

// ---------------------------------------------------------------------------
// 2) s_j = sqrt(1 + distinct-in-degree)   (diagonal contributes the +1)
__global__ void degree_kernel(const int* __restrict__ colcnt,
                              float* __restrict__ sdeg) {
    int t = blockIdx.x * blockDim.x + threadIdx.x;
    if (t < ROWS) sdeg[t] = sqrtf(1.0f + (float)colcnt[t]);
}

// ---------------------------------------------------------------------------
// 3) Y = E @ W^T  via V_WMMA_F32_16X16X4_F32.
//    Block = 128 threads (4 waves). Each wave owns a 16-row tile and all 8
//    16-col tiles (8 x v8f accumulators).
//    W is staged in LDS as K-pair-major float2 so each B fragment is a single
//    aligned ds_load_b64 into a contiguous VGPR pair (no mov shuffles):
//      BP[kp*128 + n] = ( W[n][2kp], W[n][2kp+1] )  ==  ( B[2kp][n], B[2kp+1][n] )
__global__ void __launch_bounds__(128)
gemm_ewt_kernel(const float* __restrict__ E, const float* __restrict__ W,
                float* __restrict__ Y) {
    __shared__ v2f BP[(Dn / 2) * Dn];          // 64 KB
    int tid = threadIdx.x;
    // Fill: i -> (d, kp); global read of W[d][2kp..2kp+1] is a contiguous float2.
    for (int i = tid; i < (Dn / 2) * Dn; i += blockDim.x) {
        int d  = i >> 6;            // 0..127
        int kp = i & 63;            // 0..63
        v2f w = *(const v2f*)(W + (size_t)d * Dn + 2 * kp);
        BP[kp * Dn + d] = w;
    }
    __syncthreads();

    int wave = tid >> 5;
    int lane = tid & 31;
    int half = lane >> 4;             // 0: K pair (k0,k0+1); 1: (k0+2,k0+3)
    int lr   = lane & 15;
    int rowBase = (blockIdx.x * 4 + wave) * 16;
    const float* Arow = E + (size_t)(rowBase + lr) * Dn + half * 2;

    v8f acc[8] = {};
    for (int k0 = 0; k0 < Dn; k0 += 4) {
        // A frag: one global_load_b64 (K = k0+2*half, k0+2*half+1)
        v2f a = *(const v2f*)(Arow + k0);
        const v2f* Bk = BP + ((k0 >> 1) + half) * Dn;
#pragma unroll
        for (int ct = 0; ct < 8; ++ct) {
            v2f bfrag = Bk[ct * 16 + lr];      // one ds_load_b64
            acc[ct] = __builtin_amdgcn_wmma_f32_16x16x4_f32(
                false, a, false, bfrag, (short)0, acc[ct], false, false);
        }
    }
#pragma unroll
    for (int ct = 0; ct < 8; ++ct) {
        int col = ct * 16 + lr;
#pragma unroll
        for (int v = 0; v < 8; ++v) {
            int row = rowBase + v + half * 8;      // C/D layout: M = v (+8)
            Y[(size_t)row * Dn + col] = acc[ct][v];
        }
    }
}

// ---------------------------------------------------------------------------
// 4) Eg = relu( s_i * (s_i*Y_i + sum_{edges i->j} s_j*Y_j) + b )
//    One wave per row; lane owns a float4 slice of D=128.
__global__ void gcn_agg_kernel(const float* __restrict__ Y,
                               const unsigned* __restrict__ mask,
                               const float* __restrict__ sdeg,
                               const float* __restrict__ bvec,
                               float* __restrict__ Eg) {
    int wave = threadIdx.x >> 5;
    int lane = threadIdx.x & 31;
    int row  = blockIdx.x * 8 + wave;
    if (row >= ROWS) return;
    int b = row >> 12;
    int cbase = b << 12;

    const float4* Y4 = (const float4*)Y;
    float  si = sdeg[row];
    float4 y  = Y4[(size_t)row * 32 + lane];
    float4 acc = make_float4(si * y.x, si * y.y, si * y.z, si * y.w);

    const unsigned* mrow = mask + (size_t)row * MASK_WORDS;
    for (int w = 0; w < MASK_WORDS; ++w) {
        unsigned word = mrow[w];
        while (word) {
            int j = (w << 5) + __builtin_ctz(word);
            word &= word - 1;
            float  sj = sdeg[cbase + j];
            float4 yj = Y4[(size_t)(cbase + j) * 32 + lane];
            acc.x += sj * yj.x; acc.y += sj * yj.y;
            acc.z += sj * yj.z; acc.w += sj * yj.w;
        }
    }
    float4 bb = ((const float4*)bvec)[lane];
    float4 o;
    o.x = fmaxf(si * acc.x + bb.x, 0.0f);
    o.y = fmaxf(si * acc.y + bb.y, 0.0f);
    o.z = fmaxf(si * acc.z + bb.z, 0.0f);
    o.w = fmaxf(si * acc.w + bb.w, 0.0f);
    ((float4*)Eg)[(size_t)row * 32 + lane] = o;
}

// ---------------------------------------------------------------------------
// 5) Temporal embedding table: te[s][d], d in [0,256)
__global__ void te_kernel(float* __restrict__ te) {
    int t = blockIdx.x * blockDim.x + threadIdx.x;
    if (t >= Sn * 256) return;
    int s = t >> 8;
    int d = t & 255;
    int k = d >> 1;
    // 10000^(2k/256) = exp(k * ln(1e4)/128)
    float ang = (float)s * expf(-(float)k * (9.210340371976184f / 128.0f));
    te[t] = (d & 1) ? cosf(ang) : sinf(ang);
}

// ---------------------------------------------------------------------------
// 6) Gather + concat + temporal add. One wave per token; lane holds a float4
//    of each 128-wide half. Hist tokens first, then current-trajectory tokens.
__global__ void gather_kernel(const float* __restrict__ Eg,
                              const float* __restrict__ E,
                              const int* __restrict__ history,
                              const int* __restrict__ cur,
                              const float* __restrict__ te,
                              float* __restrict__ out) {
    int wave = threadIdx.x >> 5;
    int lane = threadIdx.x & 31;
    int t = blockIdx.x * 8 + wave;
    const int NH = Bn * Hn * Sn;          // 16384 history tokens
    const int NC = Bn * Sn;               // 512 current tokens
    int b, s, idx;
    size_t obase;
    if (t < NH) {
        b = t >> 11;                      // / (H*S)
        s = t & (Sn - 1);
        idx = history[t];
        obase = (size_t)t * 256;
    } else {
        int tc = t - NH;
        if (tc >= NC) return;
        b = tc >> 6;
        s = tc & (Sn - 1);
        idx = cur[tc];
        obase = (size_t)NH * 256 + (size_t)tc * 256;
    }
    size_t row = (size_t)(b * Cn + idx);
    const float4* Eg4 = (const float4*)Eg;
    const float4* E4  = (const float4*)E;
    const float4* te4 = (const float4*)te;
    float4 g  = Eg4[row * 32 + lane];
    float4 e  = E4[row * 32 + lane];
    float4 t1 = te4[(size_t)s * 64 + lane];
    float4 t2 = te4[(size_t)s * 64 + 32 + lane];
    float4* o = (float4*)(out + obase);
    o[lane]      = make_float4(g.x + t1.x, g.y + t1.y, g.z + t1.z, g.w + t1.w);
    o[32 + lane] = make_float4(e.x + t2.x, e.y + t2.y, e.z + t2.z, e.w + t2.w);
}

// ---------------------------------------------------------------------------
extern "C" void kernel_launch(void* const* d_in, const int* in_sizes, int n_in,
                              void* d_out, int out_size, void* d_ws, size_t ws_size,
                              hipStream_t stream) {
    (void)in_sizes; (void)n_in; (void)out_size; (void)ws_size;
    // d_in[0] = candidate_size (scalar int, == 4096)
    const int*   history = (const int*)d_in[1];   // (B,H,S)
    const int*   cur     = (const int*)d_in[2];   // (B,S)
    const float* E       = (const float*)d_in[3]; // (B,C,D)
    const float* W       = (const float*)d_in[4]; // (D,D)
    const float* bvec    = (const float*)d_in[5]; // (D,)
    float* out = (float*)d_out;

    // Workspace layout (bytes):
    //   [0,           16MB)        adjacency bitmask (B*C x 128 words)
    //   [16MB,        16MB+128KB)  colcnt (int, B*C)
    //   [16MB+128KB,  16MB+256KB)  sdeg (float, B*C)
    //   [16MB+256KB,  32MB+256KB)  Y = E@W^T
    //   [32MB+256KB,  48MB+256KB)  Eg
    //   [48MB+256KB,  48MB+320KB)  te table (S x 256 floats)
    char* ws = (char*)d_ws;
    unsigned* mask   = (unsigned*)(ws);
    int*      colcnt = (int*)   (ws + (size_t)(16 << 20));
    float*    sdeg   = (float*) (ws + (size_t)(16 << 20) + (128 << 10));
    float*    Y      = (float*) (ws + (size_t)(16 << 20) + (256 << 10));
    float*    Eg     = (float*) (ws + (size_t)(32 << 20) + (256 << 10));
    float*    te     = (float*) (ws + (size_t)(48 << 20) + (256 << 10));

    // mask + colcnt are adjacent -> one stream-ordered memset (graph-capturable)
    hipMemsetAsync(mask, 0, (size_t)(16 << 20) + (128 << 10), stream);

    const int NT = Bn * Hn * (Sn - 1);                   // 16128 transitions
    build_adj_kernel<<<(NT + 255) / 256, 256, 0, stream>>>(history, mask, colcnt);
    degree_kernel<<<ROWS / 256, 256, 0, stream>>>(colcnt, sdeg);
    gemm_ewt_kernel<<<ROWS / 64, 128, 0, stream>>>(E, W, Y);
    gcn_agg_kernel<<<ROWS / 8, 256, 0, stream>>>(Y, mask, sdeg, bvec, Eg);
    te_kernel<<<(Sn * 256 + 255) / 256, 256, 0, stream>>>(te);

    const int tokens = Bn * Hn * Sn + Bn * Sn;           // 16896 -> 2112 blocks
    gather_kernel<<<tokens / 8, 256, 0, stream>>>(Eg, E, history, cur, te, out);
}